// NeuralSurfaceReconstructor_652835029163
// MI455X (gfx1250) — compile-verified
//
#include <hip/hip_runtime.h>
#include <math.h>

#define R_RAYS 8192
#define NSTEP  50
#define NPAD   64      // per-ray points padded to 64 (weights 0 beyond 50)
#define HID    64
#define FGN    192
#define BGN    128

typedef float v2f __attribute__((ext_vector_type(2)));
typedef float v8f __attribute__((ext_vector_type(8)));

// ---------------------------------------------------------------------------
// Trilinear sample, border padding, align_corners=True. q already in [-1,1].
// vol layout: [C, S, S, S], z fastest.
// ---------------------------------------------------------------------------
__device__ __forceinline__ void trilerp(const float* __restrict__ vol, int S, int C,
                                        float qx, float qy, float qz, float* out) {
  const float sf = (float)(S - 1);
  float cx = fminf(fmaxf((qx + 1.0f) * 0.5f * sf, 0.0f), sf);
  float cy = fminf(fmaxf((qy + 1.0f) * 0.5f * sf, 0.0f), sf);
  float cz = fminf(fmaxf((qz + 1.0f) * 0.5f * sf, 0.0f), sf);
  float fx0 = floorf(cx), fy0 = floorf(cy), fz0 = floorf(cz);
  int ix0 = (int)fx0, iy0 = (int)fy0, iz0 = (int)fz0;
  float fx = cx - fx0, fy = cy - fy0, fz = cz - fz0;
  int ix1 = min(ix0 + 1, S - 1);
  int iy1 = min(iy0 + 1, S - 1);
  int iz1 = min(iz0 + 1, S - 1);
  size_t sYZ = (size_t)S * (size_t)S;
  size_t vsz = (size_t)S * sYZ;
  size_t bx0 = (size_t)ix0 * sYZ, bx1 = (size_t)ix1 * sYZ;
  size_t by0 = (size_t)iy0 * S,   by1 = (size_t)iy1 * S;
  size_t o000 = bx0 + by0 + iz0, o001 = bx0 + by0 + iz1;
  size_t o010 = bx0 + by1 + iz0, o011 = bx0 + by1 + iz1;
  size_t o100 = bx1 + by0 + iz0, o101 = bx1 + by0 + iz1;
  size_t o110 = bx1 + by1 + iz0, o111 = bx1 + by1 + iz1;
  float wx0 = 1.0f - fx, wy0 = 1.0f - fy, wz0 = 1.0f - fz;
  float w000 = wx0 * wy0 * wz0, w001 = wx0 * wy0 * fz;
  float w010 = wx0 * fy  * wz0, w011 = wx0 * fy  * fz;
  float w100 = fx  * wy0 * wz0, w101 = fx  * wy0 * fz;
  float w110 = fx  * fy  * wz0, w111 = fx  * fy  * fz;
  for (int c = 0; c < C; ++c) {
    const float* vc = vol + (size_t)c * vsz;
    out[c] = vc[o000] * w000 + vc[o001] * w001 +
             vc[o010] * w010 + vc[o011] * w011 +
             vc[o100] * w100 + vc[o101] * w101 +
             vc[o110] * w110 + vc[o111] * w111;
  }
}

// ---------------------------------------------------------------------------
// Kernel 1: per-point sector routing + trilinear gather.
//   sig[r*64+n]  = sigmoid(S(x_{r,n}))   for n < 51
//   feat[(r*64+n)*4 + {0,1,2}] = feat(x_{r,n}) for n < 50, row zeroed otherwise
//   (4th lane of feat is the K-pad for the 16x16x4 WMMA A operand)
// ---------------------------------------------------------------------------
__global__ __launch_bounds__(256) void point_kernel(
    const float* __restrict__ x,
    const float* __restrict__ fg_sdf, const float* __restrict__ fg_feat,
    const float* __restrict__ bg_sdf, const float* __restrict__ bg_feat,
    float* __restrict__ sig, float* __restrict__ feat) {
  int tid = blockIdx.x * blockDim.x + threadIdx.x;
  if (tid >= R_RAYS * NPAD) return;
  int r = tid >> 6;
  int n = tid & 63;
  float4 fr = make_float4(0.0f, 0.0f, 0.0f, 0.0f);
  if (n < NSTEP + 1) {
    const float* xp = x + ((size_t)r * (NSTEP + 1) + n) * 3;
    float px = xp[0], py = xp[1], pz = xp[2];
    bool in1 = (fabsf(px) < 1.0f) & (fabsf(py) < 1.0f) & (fabsf(pz) < 1.0f);
    bool in4 = (fabsf(px) < 4.0f) & (fabsf(py) < 4.0f) & (fabsf(pz) < 4.0f);
    bool is_f = in1;
    bool is_b = in4 && !in1;
    float sdf = 1.0f;
    if (is_f)      trilerp(fg_sdf, FGN, 1, px, py, pz, &sdf);
    else if (is_b) trilerp(bg_sdf, BGN, 1, px * 0.25f, py * 0.25f, pz * 0.25f, &sdf);
    sig[tid] = 1.0f / (1.0f + expf(-sdf));
    if (n < NSTEP) {
      float f3[3] = {0.5f, 0.5f, 0.5f};
      if (is_f)      trilerp(fg_feat, FGN, 3, px, py, pz, f3);
      else if (is_b) trilerp(bg_feat, BGN, 3, px * 0.25f, py * 0.25f, pz * 0.25f, f3);
      fr.x = f3[0]; fr.y = f3[1]; fr.z = f3[2];
    }
  }
  *(float4*)(feat + (size_t)tid * 4) = fr;
}

// ---------------------------------------------------------------------------
// Kernel 2: per-ray transmittance scan.  w[r,n] = T_n * alpha_n (n<50, else 0)
// wsum[r] = sum_n w[r,n]
// ---------------------------------------------------------------------------
__global__ __launch_bounds__(256) void ray_kernel(
    const float* __restrict__ sig, float* __restrict__ w,
    float* __restrict__ wsum) {
  int r = blockIdx.x * blockDim.x + threadIdx.x;
  if (r >= R_RAYS) return;
  const float* s = sig + (size_t)r * NPAD;
  float* wr = w + (size_t)r * NPAD;
  float T = 1.0f, acc = 0.0f;
  float sp = s[0];
  #pragma unroll 5
  for (int n = 0; n < NSTEP; ++n) {
    float sn = s[n + 1];
    float alpha = fmaxf(0.0f, (sp - sn) / sp);
    float wn = T * alpha;
    wr[n] = wn;
    acc += wn;
    T *= (1.0f - alpha);
    sp = sn;
  }
  #pragma unroll
  for (int n = NSTEP; n < NPAD; ++n) wr[n] = 0.0f;
  wsum[r] = acc;
}

// ---------------------------------------------------------------------------
// Kernel 3: one wave per ray. WMMA f32 16x16x4 input layer (bias via C),
// relu + per-point weight, column reduction, folded 64->3 output layer.
// ---------------------------------------------------------------------------
__global__ __launch_bounds__(256) void mlp_kernel(
    const float* __restrict__ feat, const float* __restrict__ w,
    const float* __restrict__ wsum,
    const float* __restrict__ w1, const float* __restrict__ b1,
    const float* __restrict__ w2, const float* __restrict__ b2,
    float* __restrict__ out) {
  int gtid = blockIdx.x * blockDim.x + threadIdx.x;
  int ray  = gtid >> 5;          // one wave32 per ray
  int lane = threadIdx.x & 31;
  int half = lane >> 4;          // 0: K rows {0,1}, 1: K rows {2,3}
  int n16  = lane & 15;

  // B operand (W1 chunk 4x16) per N-chunk. K row 3 is the zero pad.
  v2f B[4];
  float b1v[4];
  #pragma unroll
  for (int j = 0; j < 4; ++j) {
    int n = j * 16 + n16;
    B[j].x = half ? w1[2 * HID + n] : w1[0 * HID + n];
    B[j].y = half ? 0.0f            : w1[1 * HID + n];
    b1v[j] = b1[n];
  }

  const float* fbase = feat + (size_t)ray * NPAD * 4;
  const float* wbase = w + (size_t)ray * NPAD;

  float hacc[4] = {0.0f, 0.0f, 0.0f, 0.0f};
  #pragma unroll
  for (int t = 0; t < 4; ++t) {
    // A operand: lane holds point M=n16 of this tile, K pair selected by half.
    v2f A = *(const v2f*)(fbase + (size_t)(t * 16 + n16) * 4 + half * 2);
    // Per-point weights for the 8 M-rows this lane contributes to.
    const float* wp = wbase + t * 16 + half * 8;
    float4 wa = *(const float4*)(wp);
    float4 wb = *(const float4*)(wp + 4);
    float w8[8] = {wa.x, wa.y, wa.z, wa.w, wb.x, wb.y, wb.z, wb.w};
    #pragma unroll
    for (int j = 0; j < 4; ++j) {
      v8f C;
      #pragma unroll
      for (int v = 0; v < 8; ++v) C[v] = b1v[j];   // bias: constant per column
      v8f D = __builtin_amdgcn_wmma_f32_16x16x4_f32(
          false, A, false, B[j], (short)0, C, false, false);
      #pragma unroll
      for (int v = 0; v < 8; ++v)
        hacc[j] += fmaxf(D[v], 0.0f) * w8[v];
    }
  }

  // Fold the two M-halves: lanes l and l^16 hold the same column N.
  #pragma unroll
  for (int j = 0; j < 4; ++j)
    hacc[j] += __shfl_xor(hacc[j], 16, 32);

  // Output layer: C_c = sum_k s[k] * W2[k][c] + wsum * b2[c]
  float p0 = 0.0f, p1 = 0.0f, p2 = 0.0f;
  #pragma unroll
  for (int j = 0; j < 4; ++j) {
    const float* r2 = w2 + (size_t)(j * 16 + n16) * 3;
    p0 += hacc[j] * r2[0];
    p1 += hacc[j] * r2[1];
    p2 += hacc[j] * r2[2];
  }
  #pragma unroll
  for (int m = 8; m >= 1; m >>= 1) {   // reduce within the 16-lane group only
    p0 += __shfl_xor(p0, m, 32);
    p1 += __shfl_xor(p1, m, 32);
    p2 += __shfl_xor(p2, m, 32);
  }
  if (lane == 0) {
    float s = wsum[ray];
    out[(size_t)ray * 3 + 0] = p0 + s * b2[0];
    out[(size_t)ray * 3 + 1] = p1 + s * b2[1];
    out[(size_t)ray * 3 + 2] = p2 + s * b2[2];
  }
}

// ---------------------------------------------------------------------------
extern "C" void kernel_launch(void* const* d_in, const int* in_sizes, int n_in,
                              void* d_out, int out_size, void* d_ws, size_t ws_size,
                              hipStream_t stream) {
  const float* x       = (const float*)d_in[0];
  // d_in[1] = v (unused by the reference MLP)
  const float* fg_sdf  = (const float*)d_in[2];
  const float* fg_feat = (const float*)d_in[3];
  const float* bg_sdf  = (const float*)d_in[4];
  const float* bg_feat = (const float*)d_in[5];
  const float* w1      = (const float*)d_in[6];
  const float* b1      = (const float*)d_in[7];
  const float* w2      = (const float*)d_in[8];
  const float* b2      = (const float*)d_in[9];
  float* out = (float*)d_out;

  // Workspace layout (floats), all 16B aligned:
  //   feat : R*64*4   (8 MB)   sig : R*64 (2 MB)   w : R*64 (2 MB)  wsum : R
  char* ws = (char*)d_ws;
  float* feat = (float*)ws;
  float* sig  = (float*)(ws + (size_t)R_RAYS * NPAD * 4 * sizeof(float));
  float* wbuf = sig + (size_t)R_RAYS * NPAD;
  float* wsm  = wbuf + (size_t)R_RAYS * NPAD;

  point_kernel<<<(R_RAYS * NPAD) / 256, 256, 0, stream>>>(
      x, fg_sdf, fg_feat, bg_sdf, bg_feat, sig, feat);
  ray_kernel<<<R_RAYS / 256, 256, 0, stream>>>(sig, wbuf, wsm);
  mlp_kernel<<<(R_RAYS * 32) / 256, 256, 0, stream>>>(
      feat, wbuf, wsm, w1, b1, w2, b2, out);
}